// dgl_GAT_10282151707714
// MI455X (gfx1250) — compile-verified
//
#include <hip/hip_runtime.h>
#include <hip/hip_bf16.h>

typedef float v2f __attribute__((ext_vector_type(2)));
typedef float v8f __attribute__((ext_vector_type(8)));

#define X_PITCH 132     // 128 + 4 pad (even -> 8B-aligned pairs)
#define NEG_SLOPE 0.2f

// ---------------------------------------------------------------------------
// GEMM: F[N,OutF] = X[N,K] @ W[K,OutF] via V_WMMA_F32_16X16X4_F32.
// One block per 16-row tile. X tile AND transposed W staged in LDS
// (W zero-padded to Opad = ctiles*16 columns) so the k-loop is branch-free:
// per step = 2x ds_load_b64 + 1x v_wmma.  K in {64,128}, OutF in {40,64,128}.
// ---------------------------------------------------------------------------
__global__ void gat_gemm_wmma(const float* __restrict__ X,
                              const float* __restrict__ W,
                              float* __restrict__ F,
                              int N, int K, int OutF, int Opad) {
    __shared__ float sx[16 * X_PITCH];      //  8.4 KB
    __shared__ float sw[128 * 130];         // 66.6 KB (Wt, [n][k], pitch K+2)
    const int Kp   = K + 2;                 // even pitch -> 8B alignment holds
    const int row0 = blockIdx.x * 16;
    const int tid  = threadIdx.x;

    // stage X tile [16 x K]
    for (int i = tid; i < 16 * K; i += blockDim.x) {
        int r = i / K, c = i - r * K;
        float v = 0.0f;
        if (row0 + r < N) v = X[(size_t)(row0 + r) * K + c];
        sx[r * X_PITCH + c] = v;
    }
    // stage W transposed [Opad x K], zero-padded columns
    for (int i = tid; i < Opad * K; i += blockDim.x) {
        int n = i / K, k = i - n * K;
        sw[n * Kp + k] = (n < OutF) ? W[(size_t)k * OutF + n] : 0.0f;
    }
    __syncthreads();

    const int wave  = tid >> 5;
    const int lane  = tid & 31;
    const int nl    = wave * 16 + (lane & 15);  // output column (< Opad)
    const int khalf = (lane >> 4) * 2;          // lanes 0-15: K={0,1}; 16-31: K={2,3}
    const int mrow  = lane & 15;                // A row

    const float* swl = &sw[nl * Kp];
    const float* sxl = &sx[mrow * X_PITCH];

    v8f c = {0.f, 0.f, 0.f, 0.f, 0.f, 0.f, 0.f, 0.f};
    for (int kk = 0; kk < K; kk += 4) {
        v2f a = *(const v2f*)&sxl[kk + khalf];   // ds_load_b64
        v2f b = *(const v2f*)&swl[kk + khalf];   // ds_load_b64
        c = __builtin_amdgcn_wmma_f32_16x16x4_f32(false, a, false, b,
                                                  (short)0, c, false, false);
    }

    // C/D layout: VGPR i -> row i (lanes 0-15) or row 8+i (lanes 16-31)
    if (nl < OutF) {
        const int mbase = (lane >> 4) * 8;
        for (int i = 0; i < 8; ++i) {
            int m = row0 + mbase + i;
            if (m < N) F[(size_t)m * OutF + nl] = c[i];
        }
    }
}

// ---------------------------------------------------------------------------
// Per-(node, head) attention scores: el = <feat, al>, er = <feat, ar>
// ---------------------------------------------------------------------------
__global__ void gat_scores(const float* __restrict__ F,
                           const float* __restrict__ al,
                           const float* __restrict__ ar,
                           float* __restrict__ el, float* __restrict__ er,
                           int N, int H, int D) {
    int idx = blockIdx.x * blockDim.x + threadIdx.x;
    if (idx >= N * H) return;
    int n = idx / H, h = idx - n * H;
    const float* f = F + (size_t)n * H * D + (size_t)h * D;
    float sl = 0.f, sr = 0.f;
    for (int d = 0; d < D; ++d) {
        float v = f[d];
        sl = fmaf(v, al[h * D + d], sl);
        sr = fmaf(v, ar[h * D + d], sr);
    }
    el[idx] = sl;
    er[idx] = sr;
}

__global__ void fill_f32(float* __restrict__ p, size_t n, float v) {
    size_t i = (size_t)blockIdx.x * blockDim.x + threadIdx.x;
    if (i < n) p[i] = v;
}

__device__ inline void atomicMaxF32(float* addr, float val) {
    int* a = (int*)addr;
    int old = __float_as_int(*addr);
    while (__int_as_float(old) < val) {
        int prev = atomicCAS(a, old, __float_as_int(val));
        if (prev == old) break;
        old = prev;
    }
}

__device__ inline float leaky(float x) { return x > 0.f ? x : NEG_SLOPE * x; }

// pass 1: segment max of leaky_relu(el[src]+er[dst]) over dst
__global__ void edge_max(const int* __restrict__ src, const int* __restrict__ dst,
                         const float* __restrict__ el, const float* __restrict__ er,
                         float* __restrict__ m, int E, int H) {
    int i = blockIdx.x * blockDim.x + threadIdx.x;
    if (i >= E * H) return;
    int e = i / H, h = i - e * H;
    int sn = src[e], dn = dst[e];
    float x = leaky(el[sn * H + h] + er[dn * H + h]);
    atomicMaxF32(&m[dn * H + h], x);
}

// pass 2: ee = exp(e - m[dst]); segment sum over dst
__global__ void edge_expsum(const int* __restrict__ src, const int* __restrict__ dst,
                            const float* __restrict__ el, const float* __restrict__ er,
                            const float* __restrict__ m, float* __restrict__ ssum,
                            float* __restrict__ ee, int E, int H) {
    int i = blockIdx.x * blockDim.x + threadIdx.x;
    if (i >= E * H) return;
    int e = i / H, h = i - e * H;
    int sn = src[e], dn = dst[e];
    float x  = leaky(el[sn * H + h] + er[dn * H + h]);
    float ex = __expf(x - m[dn * H + h]);
    ee[i] = ex;
    atomicAdd(&ssum[dn * H + h], ex);
}

// pass 3: agg[dst] += feat[src] * (ee / s[dst])
__global__ void edge_aggregate(const int* __restrict__ src, const int* __restrict__ dst,
                               const float* __restrict__ F, const float* __restrict__ ee,
                               const float* __restrict__ ssum, float* __restrict__ agg,
                               int E, int H, int D) {
    int i = blockIdx.x * blockDim.x + threadIdx.x;
    if (i >= E * H) return;
    int e = i / H, h = i - e * H;
    int sn = src[e], dn = dst[e];
    const float* fs = F + (size_t)sn * H * D + (size_t)h * D;
    __builtin_prefetch(fs, 0, 1);                 // global_prefetch_b8: hide gather latency
    float alpha = ee[i] / ssum[dn * H + h];
    float* ad = agg + (size_t)dn * H * D + (size_t)h * D;
    for (int d = 0; d < D; ++d) atomicAdd(&ad[d], fs[d] * alpha);
}

// out = agg + bias (layer0 flatten and layer1 mean-over-1-head are identity)
__global__ void bias_add(const float* __restrict__ A, const float* __restrict__ b,
                         float* __restrict__ Y, size_t total, int Fdim) {
    size_t i = (size_t)blockIdx.x * blockDim.x + threadIdx.x;
    if (i >= total) return;
    int j = (int)(i % (size_t)Fdim);
    Y[i] = A[i] + b[j];
}

// ---------------------------------------------------------------------------
static void run_layer(const float* X, const float* W, const float* al, const float* ar,
                      const float* b, const int* src, const int* dst,
                      float* FEAT, float* AGG, float* MM, float* SS,
                      float* EL, float* ER, float* EE, float* OUT,
                      int N, int E, int K, int H, int D, hipStream_t stream) {
    const int OutF   = H * D;
    const int ctiles = (OutF + 15) / 16;
    const int Opad   = ctiles * 16;

    dim3 gridG((N + 15) / 16), blockG(32 * ctiles);
    gat_gemm_wmma<<<gridG, blockG, 0, stream>>>(X, W, FEAT, N, K, OutF, Opad);

    int tnh = N * H;
    gat_scores<<<(tnh + 255) / 256, 256, 0, stream>>>(FEAT, al, ar, EL, ER, N, H, D);

    fill_f32<<<(tnh + 255) / 256, 256, 0, stream>>>(MM, (size_t)tnh, -3.0e38f);
    fill_f32<<<(tnh + 255) / 256, 256, 0, stream>>>(SS, (size_t)tnh, 0.0f);
    size_t tagg = (size_t)N * OutF;
    fill_f32<<<(unsigned)((tagg + 255) / 256), 256, 0, stream>>>(AGG, tagg, 0.0f);

    int teh = E * H;
    edge_max<<<(teh + 255) / 256, 256, 0, stream>>>(src, dst, EL, ER, MM, E, H);
    edge_expsum<<<(teh + 255) / 256, 256, 0, stream>>>(src, dst, EL, ER, MM, SS, EE, E, H);
    edge_aggregate<<<(teh + 255) / 256, 256, 0, stream>>>(src, dst, FEAT, EE, SS, AGG, E, H, D);

    bias_add<<<(unsigned)((tagg + 255) / 256), 256, 0, stream>>>(AGG, b, OUT, tagg, OutF);
}

extern "C" void kernel_launch(void* const* d_in, const int* in_sizes, int n_in,
                              void* d_out, int out_size, void* d_ws, size_t ws_size,
                              hipStream_t stream) {
    const float* in_feat = (const float*)d_in[0];
    const int*   src     = (const int*)d_in[1];
    const int*   dst     = (const int*)d_in[2];
    const float* W0  = (const float*)d_in[3];
    const float* al0 = (const float*)d_in[4];
    const float* ar0 = (const float*)d_in[5];
    const float* b0  = (const float*)d_in[6];
    const float* W1  = (const float*)d_in[7];
    const float* al1 = (const float*)d_in[8];
    const float* ar1 = (const float*)d_in[9];
    const float* b1  = (const float*)d_in[10];
    const float* W2  = (const float*)d_in[11];
    const float* al2 = (const float*)d_in[12];
    const float* ar2 = (const float*)d_in[13];
    const float* b2  = (const float*)d_in[14];

    const int IN = 128, HID = 64, OUTF = 40;
    const int N = in_sizes[0] / IN;
    const int E = in_sizes[1];

    // workspace carving (floats)
    float* ws   = (float*)d_ws;
    size_t nf   = (size_t)N * 128;
    float* FEAT = ws;                 // N*128
    float* AGG  = ws + nf;            // N*128
    float* XBUF = ws + 2 * nf;        // N*128 (inter-layer activations)
    float* EL   = ws + 3 * nf;        // N*2
    float* ER   = EL + (size_t)N * 2;
    float* MM   = ER + (size_t)N * 2;
    float* SS   = MM + (size_t)N * 2;
    float* EE   = SS + (size_t)N * 2; // E*2

    // layer 0: [N,128] -> heads=2, D=64 -> flatten [N,128]
    run_layer(in_feat, W0, al0, ar0, b0, src, dst,
              FEAT, AGG, MM, SS, EL, ER, EE, XBUF,
              N, E, IN, 2, HID, stream);

    // layer 1: [N,128] -> heads=1, D=64 -> mean over heads (H=1) [N,64]
    run_layer(XBUF, W1, al1, ar1, b1, src, dst,
              FEAT, AGG, MM, SS, EL, ER, EE, XBUF,
              N, E, 2 * HID, 1, HID, stream);

    // layer 2: [N,64] -> heads=1, D=40 -> flatten [N,40] into d_out
    run_layer(XBUF, W2, al2, ar2, b2, src, dst,
              FEAT, AGG, MM, SS, EL, ER, EE, (float*)d_out,
              N, E, HID, 1, OUTF, stream);
}